// LoRALinearExpertsWrapper_21552145891711
// MI455X (gfx1250) — compile-verified
//
#include <hip/hip_runtime.h>
#include <hip/hip_bf16.h>
#include <cstdint>

// ---------------------------------------------------------------------------
// LoRA SwiGLU expert MLP for MI455X (gfx1250, wave32, WMMA).
// - All GEMMs on V_WMMA_F32_16X16X4_F32 (exact fp32, matches reference).
// - LoRA rank-64 terms folded into the main GEMMs as a K-dimension tail.
// - SwiGLU fused into the gate/up epilogue; gate_up never touches HBM.
// - Global->LDS staging via GLOBAL_LOAD_ASYNC_TO_LDS_B128 (ASYNCcnt) with
//   double-buffered LDS: async copies for slab k+1 overlap WMMA on slab k.
// ---------------------------------------------------------------------------

#define TOK        8192   // NUM_EXPERTS * TOKENS_PER_EXPERT
#define HIDDEN     2048
#define EXPERT_DIM 4096
#define RANK       64
#define SCALING    0.25f  // LORA_ALPHA / R = 16 / 64

typedef __attribute__((ext_vector_type(2))) float v2f;
typedef __attribute__((ext_vector_type(8))) float v8f;
typedef __attribute__((ext_vector_type(4))) int   v4i;

#if defined(__gfx1250__) && \
    __has_builtin(__builtin_amdgcn_global_load_async_to_lds_b128) && \
    __has_builtin(__builtin_amdgcn_s_wait_asynccnt)
#define HAVE_ASYNC_LDS 1
#else
#define HAVE_ASYNC_LDS 0
#endif

#if HAVE_ASYNC_LDS
// Builtin prototype (per hipcc diagnostic): (AS1 int4*, AS3 int4*, imm, imm)
typedef __attribute__((address_space(1))) v4i gv4i;  // global int4
typedef __attribute__((address_space(3))) v4i lv4i;  // LDS int4
#endif

// D(16x16,f32) = A(16x4,f32) x B(4x16,f32) + C
__device__ __forceinline__ v8f wmma4(v2f a, v2f b, v8f c) {
  return __builtin_amdgcn_wmma_f32_16x16x4_f32(
      /*neg_a=*/false, a, /*neg_b=*/false, b,
      /*c_mod=*/(short)0, c, /*reuse_a=*/false, /*reuse_b=*/false);
}

// LDS row stride: 32 payload + 4 pad floats. 144B rows keep b128 stores
// 16B-aligned and make the b64 fragment reads bank-conflict free
// (36*m mod 64 hits 16 distinct banks; x4 dwords covers all 64 banks).
#define LDS_STRIDE 36

// --- Tile staging -----------------------------------------------------------
// Stage an nrows x 32 fp32 tile (row stride ld, col offset k0) into LDS.
// Async path: per-lane 16B copies, VGPRs bypassed, tracked by ASYNCcnt.
__device__ __forceinline__ void stage_tile(const float* __restrict__ G,
                                           int row0, long ld, int k0,
                                           float* __restrict__ S, int nrows) {
  const int total = nrows * 8;  // float4 quanta
  for (int i = threadIdx.x; i < total; i += 256) {
    const int r = i >> 3;
    const int c = (i & 7) << 2;
    const float* gp = G + (long)(row0 + r) * ld + k0 + c;
    float* lp = S + r * LDS_STRIDE + c;
#if HAVE_ASYNC_LDS
    __builtin_amdgcn_global_load_async_to_lds_b128(
        (gv4i*)(uintptr_t)gp,                 // 64-bit global address
        (lv4i*)(uint32_t)(uintptr_t)lp,       // low 32 bits = LDS offset
        /*offset=*/0, /*cpol=*/0);
#else
    *reinterpret_cast<float4*>(lp) = *reinterpret_cast<const float4*>(gp);
#endif
  }
}

// Wave waits for its outstanding async copies, then workgroup barrier:
// after this, every wave's staged data is visible in LDS.
__device__ __forceinline__ void stage_commit() {
#if HAVE_ASYNC_LDS
  __builtin_amdgcn_s_wait_asynccnt(0);
#endif
  __syncthreads();
}

// --- WMMA slab: 8 K-chunks of 4, 2x2 tiles of 16x16, single B matrix -------
__device__ __forceinline__ void mma_2x2(const float* __restrict__ As,
                                        const float* __restrict__ Bs, int arow,
                                        int brow, int lm, int lk,
                                        v8f acc[2][2]) {
#pragma unroll
  for (int c = 0; c < 8; ++c) {
    const int kk = c * 4 + lk * 2;  // lane->K mapping for f32 WMMA frags
    const v2f a0 = *reinterpret_cast<const v2f*>(As + (arow + lm) * LDS_STRIDE + kk);
    const v2f a1 = *reinterpret_cast<const v2f*>(As + (arow + 16 + lm) * LDS_STRIDE + kk);
    const v2f b0 = *reinterpret_cast<const v2f*>(Bs + (brow + lm) * LDS_STRIDE + kk);
    const v2f b1 = *reinterpret_cast<const v2f*>(Bs + (brow + 16 + lm) * LDS_STRIDE + kk);
    acc[0][0] = wmma4(a0, b0, acc[0][0]);
    acc[0][1] = wmma4(a0, b1, acc[0][1]);
    acc[1][0] = wmma4(a1, b0, acc[1][0]);
    acc[1][1] = wmma4(a1, b1, acc[1][1]);
  }
}

// ---------------------------------------------------------------------------
// Kernel 1/3: T = SCALING * (X @ A^T).  X:(TOK,K)  A:(RANK,K)  T:(TOK,RANK)
// Block = 128 rows x all 64 cols; 8 waves in 4x2, 32x32 per wave.
// ---------------------------------------------------------------------------
__global__ __launch_bounds__(256) void lora_proj(const float* __restrict__ X,
                                                 const float* __restrict__ A,
                                                 float* __restrict__ T, int K) {
  __shared__ float xs[2][128 * LDS_STRIDE];
  __shared__ float as[2][64 * LDS_STRIDE];
  const int m0 = blockIdx.x * 128;
  const int lane = threadIdx.x & 31;
  const int wave = threadIdx.x >> 5;
  const int wm = wave >> 1, wn = wave & 1;
  const int lm = lane & 15, lk = lane >> 4;
  v8f acc[2][2] = {};
  const int NT = K / 32;
  stage_tile(X, m0, K, 0, xs[0], 128);
  stage_tile(A, 0, K, 0, as[0], 64);
  stage_commit();
  for (int kt = 0; kt < NT; ++kt) {
    const int cur = kt & 1, nxt = cur ^ 1;
    if (kt + 1 < NT) {  // prefetch slab k+1 while WMMAs run on slab k
      const int k0 = (kt + 1) * 32;
      stage_tile(X, m0, K, k0, xs[nxt], 128);
      stage_tile(A, 0, K, k0, as[nxt], 64);
    }
    mma_2x2(xs[cur], as[cur], wm * 32, wn * 32, lm, lk, acc);
    stage_commit();
  }
#pragma unroll
  for (int tm = 0; tm < 2; ++tm)
#pragma unroll
    for (int tn = 0; tn < 2; ++tn) {
      const int mm = m0 + wm * 32 + tm * 16 + lk * 8;
      const int nn = wn * 32 + tn * 16 + lm;
#pragma unroll
      for (int v = 0; v < 8; ++v)
        T[(long)(mm + v) * RANK + nn] = SCALING * acc[tm][tn][v];
    }
}

// ---------------------------------------------------------------------------
// Kernel 2: gate/up GEMM with fused LoRA K-tail and fused SwiGLU.
//   gate = X@Wgu[n]^T      + T1@Bgu[n]^T
//   up   = X@Wgu[4096+n]^T + T1@Bgu[4096+n]^T
//   H    = up * silu(gate)                      H:(TOK, EXPERT_DIM)
// A-fragments (X rows) are shared between the gate and up WMMA streams.
// ---------------------------------------------------------------------------
__device__ __forceinline__ void gateup_stage(int kt, int m0, int n0,
                                             const float* __restrict__ X,
                                             const float* __restrict__ Wgu,
                                             const float* __restrict__ T1,
                                             const float* __restrict__ Bgu,
                                             float* xs, float* wg, float* wu) {
  const int MAIN = HIDDEN / 32;
  if (kt < MAIN) {
    const int k0 = kt * 32;
    stage_tile(X, m0, HIDDEN, k0, xs, 128);
    stage_tile(Wgu, n0, HIDDEN, k0, wg, 64);
    stage_tile(Wgu, EXPERT_DIM + n0, HIDDEN, k0, wu, 64);
  } else {  // LoRA tail: K extends by RANK using [T1 | Bgu]
    const int k0 = (kt - MAIN) * 32;
    stage_tile(T1, m0, RANK, k0, xs, 128);
    stage_tile(Bgu, n0, RANK, k0, wg, 64);
    stage_tile(Bgu, EXPERT_DIM + n0, RANK, k0, wu, 64);
  }
}

__global__ __launch_bounds__(256) void gateup_swiglu(
    const float* __restrict__ X, const float* __restrict__ Wgu,
    const float* __restrict__ T1, const float* __restrict__ Bgu,
    float* __restrict__ H) {
  __shared__ float xs[2][128 * LDS_STRIDE];
  __shared__ float wg[2][64 * LDS_STRIDE];
  __shared__ float wu[2][64 * LDS_STRIDE];
  const int m0 = blockIdx.x * 128;
  const int n0 = blockIdx.y * 64;
  const int lane = threadIdx.x & 31;
  const int wave = threadIdx.x >> 5;
  const int wm = wave >> 1, wn = wave & 1;
  const int lm = lane & 15, lk = lane >> 4;
  v8f accg[2][2] = {}, accu[2][2] = {};
  const int NT = HIDDEN / 32 + RANK / 32;
  gateup_stage(0, m0, n0, X, Wgu, T1, Bgu, xs[0], wg[0], wu[0]);
  stage_commit();
  for (int kt = 0; kt < NT; ++kt) {
    const int cur = kt & 1, nxt = cur ^ 1;
    if (kt + 1 < NT)
      gateup_stage(kt + 1, m0, n0, X, Wgu, T1, Bgu, xs[nxt], wg[nxt], wu[nxt]);
#pragma unroll
    for (int c = 0; c < 8; ++c) {
      const int kk = c * 4 + lk * 2;
      const v2f a0 = *reinterpret_cast<const v2f*>(xs[cur] + (wm * 32 + lm) * LDS_STRIDE + kk);
      const v2f a1 = *reinterpret_cast<const v2f*>(xs[cur] + (wm * 32 + 16 + lm) * LDS_STRIDE + kk);
      const v2f g0 = *reinterpret_cast<const v2f*>(wg[cur] + (wn * 32 + lm) * LDS_STRIDE + kk);
      const v2f g1 = *reinterpret_cast<const v2f*>(wg[cur] + (wn * 32 + 16 + lm) * LDS_STRIDE + kk);
      const v2f u0 = *reinterpret_cast<const v2f*>(wu[cur] + (wn * 32 + lm) * LDS_STRIDE + kk);
      const v2f u1 = *reinterpret_cast<const v2f*>(wu[cur] + (wn * 32 + 16 + lm) * LDS_STRIDE + kk);
      accg[0][0] = wmma4(a0, g0, accg[0][0]);
      accg[0][1] = wmma4(a0, g1, accg[0][1]);
      accg[1][0] = wmma4(a1, g0, accg[1][0]);
      accg[1][1] = wmma4(a1, g1, accg[1][1]);
      accu[0][0] = wmma4(a0, u0, accu[0][0]);
      accu[0][1] = wmma4(a0, u1, accu[0][1]);
      accu[1][0] = wmma4(a1, u0, accu[1][0]);
      accu[1][1] = wmma4(a1, u1, accu[1][1]);
    }
    stage_commit();
  }
  // Fused SwiGLU epilogue.
#pragma unroll
  for (int tm = 0; tm < 2; ++tm)
#pragma unroll
    for (int tn = 0; tn < 2; ++tn) {
      const int mm = m0 + wm * 32 + tm * 16 + lk * 8;
      const int nn = n0 + wn * 32 + tn * 16 + lm;
#pragma unroll
      for (int v = 0; v < 8; ++v) {
        const float g = accg[tm][tn][v];
        const float u = accu[tm][tn][v];
        const float s = g / (1.0f + __expf(-g));  // silu(g)
        H[(long)(mm + v) * EXPERT_DIM + nn] = u * s;
      }
    }
}

// ---------------------------------------------------------------------------
// Kernel 4: Out = H @ Wd^T + T2 @ Bd^T   (LoRA fused as K-tail)
// ---------------------------------------------------------------------------
__global__ __launch_bounds__(256) void down_proj(const float* __restrict__ Hm,
                                                 const float* __restrict__ Wd,
                                                 const float* __restrict__ T2,
                                                 const float* __restrict__ Bd,
                                                 float* __restrict__ Out) {
  __shared__ float xs[2][128 * LDS_STRIDE];
  __shared__ float ws[2][64 * LDS_STRIDE];
  const int m0 = blockIdx.x * 128;
  const int n0 = blockIdx.y * 64;
  const int lane = threadIdx.x & 31;
  const int wave = threadIdx.x >> 5;
  const int wm = wave >> 1, wn = wave & 1;
  const int lm = lane & 15, lk = lane >> 4;
  v8f acc[2][2] = {};
  const int MAIN = EXPERT_DIM / 32, NT = MAIN + RANK / 32;
  stage_tile(Hm, m0, EXPERT_DIM, 0, xs[0], 128);
  stage_tile(Wd, n0, EXPERT_DIM, 0, ws[0], 64);
  stage_commit();
  for (int kt = 0; kt < NT; ++kt) {
    const int cur = kt & 1, nxt = cur ^ 1;
    if (kt + 1 < NT) {
      const int kn = kt + 1;
      if (kn < MAIN) {
        stage_tile(Hm, m0, EXPERT_DIM, kn * 32, xs[nxt], 128);
        stage_tile(Wd, n0, EXPERT_DIM, kn * 32, ws[nxt], 64);
      } else {
        stage_tile(T2, m0, RANK, (kn - MAIN) * 32, xs[nxt], 128);
        stage_tile(Bd, n0, RANK, (kn - MAIN) * 32, ws[nxt], 64);
      }
    }
    mma_2x2(xs[cur], ws[cur], wm * 32, wn * 32, lm, lk, acc);
    stage_commit();
  }
#pragma unroll
  for (int tm = 0; tm < 2; ++tm)
#pragma unroll
    for (int tn = 0; tn < 2; ++tn) {
      const int mm = m0 + wm * 32 + tm * 16 + lk * 8;
      const int nn = n0 + wn * 32 + tn * 16 + lm;
#pragma unroll
      for (int v = 0; v < 8; ++v)
        Out[(long)(mm + v) * HIDDEN + nn] = acc[tm][tn][v];
    }
}

// ---------------------------------------------------------------------------
extern "C" void kernel_launch(void* const* d_in, const int* in_sizes, int n_in,
                              void* d_out, int out_size, void* d_ws,
                              size_t ws_size, hipStream_t stream) {
  const float* X   = (const float*)d_in[0];  // (8192, 2048)
  const float* Wgu = (const float*)d_in[1];  // (8192, 2048)
  const float* Agu = (const float*)d_in[2];  // (64, 2048)
  const float* Bgu = (const float*)d_in[3];  // (8192, 64)
  const float* Wd  = (const float*)d_in[4];  // (2048, 4096)
  const float* Ad  = (const float*)d_in[5];  // (64, 4096)
  const float* Bd  = (const float*)d_in[6];  // (2048, 64)
  float* out = (float*)d_out;                // (8192, 2048)

  // Workspace: t1 (2MB) | t2 (2MB) | h (128MB). All fully overwritten each
  // call before being read -> deterministic under 0xAA poisoning.
  float* t1 = (float*)d_ws;                  // (TOK, RANK), pre-scaled
  float* t2 = t1 + (size_t)TOK * RANK;       // (TOK, RANK), pre-scaled
  float* h  = t2 + (size_t)TOK * RANK;       // (TOK, EXPERT_DIM)

  const dim3 blk(256);
  lora_proj<<<dim3(TOK / 128), blk, 0, stream>>>(X, Agu, t1, HIDDEN);
  gateup_swiglu<<<dim3(TOK / 128, EXPERT_DIM / 64), blk, 0, stream>>>(
      X, Wgu, t1, Bgu, h);
  lora_proj<<<dim3(TOK / 128), blk, 0, stream>>>(h, Ad, t2, EXPERT_DIM);
  down_proj<<<dim3(TOK / 128, HIDDEN / 64), blk, 0, stream>>>(h, Wd, t2, Bd,
                                                              out);
}